// HierarchicalCategoricalLogLikelihoodLossLayer_86835648790827
// MI455X (gfx1250) — compile-verified
//
#include <hip/hip_runtime.h>
#include <hip/hip_bf16.h>
#include <math.h>

#define BB 64
#define TT 512
#define DD 64
#define KK 12
#define PP 100
#define LOG2PI 1.8378770664093453f

typedef float v2f __attribute__((ext_vector_type(2)));
typedef float v8f __attribute__((ext_vector_type(8)));
typedef unsigned int u32x4 __attribute__((ext_vector_type(4)));
typedef int i32x4 __attribute__((ext_vector_type(4)));
typedef int i32x8 __attribute__((ext_vector_type(8)));

// ws layout (floats): [0]=sum(gamma*logp), [1]=mu_reg, [2]=L_reg, [16..16+PP)=subject mask
#define WS_MASK 16

// t-tile: 32 time columns per TDM transfer; LDS row stride 68 floats (64 + 4 pad)
#define TCOLS 32
#define RSTRIDE 68

__global__ __launch_bounds__(128) void hgmm_init_ws(float* ws) {
    for (int i = threadIdx.x; i < WS_MASK + PP; i += 128) ws[i] = 0.0f;
}

__global__ __launch_bounds__(128) void hgmm_mask(const int* __restrict__ sid,
                                                 float* __restrict__ ws) {
    int p = threadIdx.x;
    if (p < PP) {
        float m = 0.0f;
        for (int b = 0; b < BB; ++b)
            if (sid[b] == p) m = 1.0f;
        ws[WS_MASK + p] = m;
    }
}

// Issue a TDM 2D tile load: tile TCOLS rows x 64 cols (f32) from a 512x64 tensor,
// LDS destination padded by 4 dwords every 64 dwords (row stride 68 floats).
__device__ __forceinline__ void tdm_issue_tile(const float* gptr, unsigned lds_byte_off) {
    unsigned long long ga = (unsigned long long)(const void*)gptr;
    u32x4 g0;
    g0.x = 1u;                                            // count=1, user desc
    g0.y = lds_byte_off;                                  // lds_addr (bytes)
    g0.z = (unsigned)(ga & 0xffffffffu);                  // global_addr[31:0]
    g0.w = ((unsigned)((ga >> 32) & 0x01ffffffu)) | (2u << 30);  // addr[56:32] | type=2
    i32x8 g1;
    g1[0] = (2 << 16)      // data_size = 4B
          | (1 << 20)      // pad_enable
          | (5 << 22)      // pad_interval: 64 dwords
          | (3 << 25);     // pad_amount: 4 dwords
    g1[1] = (DD & 0xffff) << 16;     // tensor_dim0 = 64 (bits 79:48, low part)
    g1[2] = (TT & 0xffff) << 16;     // tensor_dim1 = 512 (bits 111:80, low part)
    g1[3] = (DD & 0xffff) << 16;     // tile_dim0 = 64 (bits 127:112)
    g1[4] = TCOLS;                   // tile_dim1 = 32 (bits 143:128)
    g1[5] = DD;                      // tensor_dim0_stride = 64 (bits 207:160, low)
    g1[6] = 0;
    g1[7] = 0;
    i32x4 z4 = {0, 0, 0, 0};
    i32x8 z8 = {0, 0, 0, 0, 0, 0, 0, 0};
    __builtin_amdgcn_tensor_load_to_lds(g0, g1, z4, z4, z8, 0);
}

// One block per (b,k): invert tril(L_subj[sid[b],k]) in LDS, w = Linv@mu, then
// z = Linv @ x_tile via V_WMMA_F32_16X16X4_F32 with TDM double-buffered x tiles;
// maha column sums from (z - w)^2; accumulate sum(gamma * logp) into ws[0].
__global__ __launch_bounds__(128) void hgmm_main(const float* __restrict__ x,
                                                 const float* __restrict__ mu_subj,
                                                 const float* __restrict__ L_subj,
                                                 const float* __restrict__ gamma,
                                                 const int* __restrict__ sid,
                                                 float* __restrict__ ws) {
    __shared__ float Ls[DD * 65];                         // L (tril), padded
    __shared__ float Li[DD * 65];                         // L^{-1}, padded
    __shared__ __align__(16) float tbuf0[TCOLS * RSTRIDE]; // x tile, t-major, padded
    __shared__ __align__(16) float tbuf1[TCOLS * RSTRIDE];
    __shared__ float mus[DD];
    __shared__ float wlds[DD];                            // w = Linv @ mu
    __shared__ float colsum[2][TCOLS];
    __shared__ float gred[TCOLS];
    __shared__ float logdet_s;

    const int blk = blockIdx.x;
    const int b = blk / KK;
    const int k = blk % KK;
    const int tid = threadIdx.x;
    const int wave = tid >> 5;          // wave32: 4 waves
    const int lane = tid & 31;
    const int lane16 = lane & 15;
    const int khalf = (lane < 16) ? 0 : 2;
    const int s = sid[b];

    const float* Lg  = L_subj  + (size_t)(s * KK + k) * DD * DD;
    const float* mug = mu_subj + (size_t)(s * KK + k) * DD;
    const float* xb  = x + (size_t)b * TT * DD;

    // LDS byte offsets of the two tile buffers (scalar ptrtoint; low 32 bits
    // of a generic shared pointer are the LDS offset). Kept as scalars to
    // avoid addrspacecast expressions in static initializers.
    const unsigned off0 = (unsigned)(size_t)(void*)tbuf0;
    const unsigned off1 = (unsigned)(size_t)(void*)tbuf1;

    // kick off TDM for tile 0 ASAP; overlaps with inversion below
    if (wave == 0) tdm_issue_tile(xb, off0);

    // ---- load tril(L) into LDS ----
    for (int i = tid; i < DD * DD; i += 128) {
        int r_ = i / DD, c_ = i % DD;
        float v = Lg[i];
        Ls[r_ * 65 + c_] = (c_ <= r_) ? v : 0.0f;
    }
    if (tid < DD) mus[tid] = mug[tid];
    __syncthreads();

    // ---- logdet ----
    if (tid == 0) {
        float ld = 0.0f;
        for (int i = 0; i < DD; ++i) ld += __logf(Ls[i * 65 + i]);
        logdet_s = ld;
    }

    // ---- triangular inversion: thread j owns column j (independent columns) ----
    if (tid < DD) {
        const int j = tid;
        for (int i = 0; i < j; ++i) Li[i * 65 + j] = 0.0f;
        for (int i = j; i < DD; ++i) {
            float sum = (i == j) ? 1.0f : 0.0f;
            for (int m = j; m < i; ++m)
                sum -= Ls[i * 65 + m] * Li[m * 65 + j];
            Li[i * 65 + j] = sum / Ls[i * 65 + i];
        }
    }
    __syncthreads();

    // ---- w = Linv @ mu (row dot; Linv row i has nonzeros j<=i) ----
    if (tid < DD) {
        float acc = 0.0f;
        for (int j = 0; j <= tid; ++j) acc += Li[tid * 65 + j] * mus[j];
        wlds[tid] = acc;
    }
    __syncthreads();

    // ---- hoist A fragments (Linv rows, loop-invariant over T) ----
    const int row = wave * 16 + lane16;
    v2f afrag[16];
#pragma unroll
    for (int kk = 0; kk < 16; ++kk) {
        const int kb = kk * 4 + khalf;
        afrag[kk].x = Li[row * 65 + kb];
        afrag[kk].y = Li[row * 65 + kb + 1];
    }
    float wv[8];
#pragma unroll
    for (int v = 0; v < 8; ++v)
        wv[v] = wlds[wave * 16 + v + ((lane < 16) ? 0 : 8)];

    float gsum = 0.0f;
    const int NT = TT / TCOLS;  // 16 tiles

    for (int i = 0; i < NT; ++i) {
        if (wave == 0) {
            if (i + 1 < NT) {
                tdm_issue_tile(xb + (size_t)(i + 1) * TCOLS * DD,
                               ((i + 1) & 1) ? off1 : off0);
                __builtin_amdgcn_s_wait_tensorcnt((short)1);  // tile i landed
            } else {
                __builtin_amdgcn_s_wait_tensorcnt((short)0);
            }
        }
        if (tid < TCOLS) colsum[i & 1][tid] = 0.0f;
        __syncthreads();  // tile i visible to all waves; colsum zeroed

        const float* rtT = (i & 1) ? tbuf1 : tbuf0;  // [t][d], stride RSTRIDE
        v8f acc0 = {0.f, 0.f, 0.f, 0.f, 0.f, 0.f, 0.f, 0.f};
        v8f acc1 = {0.f, 0.f, 0.f, 0.f, 0.f, 0.f, 0.f, 0.f};
#pragma unroll
        for (int kk = 0; kk < 16; ++kk) {
            const int kb = kk * 4 + khalf;
            const v2f b0 = *reinterpret_cast<const v2f*>(rtT + lane16 * RSTRIDE + kb);
            const v2f b1 = *reinterpret_cast<const v2f*>(rtT + (lane16 + 16) * RSTRIDE + kb);
            acc0 = __builtin_amdgcn_wmma_f32_16x16x4_f32(
                false, afrag[kk], false, b0, (short)0, acc0, false, false);
            acc1 = __builtin_amdgcn_wmma_f32_16x16x4_f32(
                false, afrag[kk], false, b1, (short)0, acc1, false, false);
        }

        float ss0 = 0.0f, ss1 = 0.0f;
#pragma unroll
        for (int v = 0; v < 8; ++v) {
            float y0 = acc0[v] - wv[v];
            float y1 = acc1[v] - wv[v];
            ss0 += y0 * y0;
            ss1 += y1 * y1;
        }
        atomicAdd(&colsum[i & 1][lane16], ss0);
        atomicAdd(&colsum[i & 1][lane16 + 16], ss1);
        __syncthreads();

        if (tid < TCOLS) {
            float maha = colsum[i & 1][tid];
            float logp = -0.5f * maha - logdet_s - 0.5f * (float)DD * LOG2PI;
            float g = gamma[((size_t)b * TT + (i * TCOLS + tid)) * KK + k];
            gsum += g * logp;
        }
    }

    __syncthreads();
    if (tid < TCOLS) gred[tid] = gsum;
    __syncthreads();
    if (tid == 0) {
        float tot = 0.0f;
        for (int i = 0; i < TCOLS; ++i) tot += gred[i];
        atomicAdd(&ws[0], tot);
    }
}

// Masked regularizer sums.
__global__ __launch_bounds__(256) void hgmm_reg(const float* __restrict__ mu_pop,
                                                const float* __restrict__ L_pop,
                                                const float* __restrict__ mu_subj,
                                                const float* __restrict__ L_subj,
                                                float* __restrict__ ws) {
    __shared__ float red[256];
    const int idx = blockIdx.x * 256 + threadIdx.x;

    float lacc = 0.0f, macc = 0.0f;
    if (idx < PP * KK * DD * DD) {
        int p = idx / (KK * DD * DD);
        int rem = idx % (KK * DD * DD);
        float d = L_subj[idx] - L_pop[rem];
        lacc = ws[WS_MASK + p] * d * d;
    }
    if (idx < PP * KK * DD) {
        int p = idx / (KK * DD);
        int rem = idx % (KK * DD);
        float d = mu_subj[idx] - mu_pop[rem];
        macc = ws[WS_MASK + p] * d * d;
    }

    red[threadIdx.x] = lacc;
    __syncthreads();
    for (int s = 128; s > 0; s >>= 1) {
        if (threadIdx.x < s) red[threadIdx.x] += red[threadIdx.x + s];
        __syncthreads();
    }
    if (threadIdx.x == 0 && red[0] != 0.0f) atomicAdd(&ws[2], red[0]);
    __syncthreads();

    red[threadIdx.x] = macc;
    __syncthreads();
    for (int s = 128; s > 0; s >>= 1) {
        if (threadIdx.x < s) red[threadIdx.x] += red[threadIdx.x + s];
        __syncthreads();
    }
    if (threadIdx.x == 0 && red[0] != 0.0f) atomicAdd(&ws[1], red[0]);
}

__global__ __launch_bounds__(32) void hgmm_finalize(const float* __restrict__ ws,
                                                    float* __restrict__ out) {
    if (threadIdx.x == 0 && blockIdx.x == 0) {
        float scount = 0.0f;
        for (int p = 0; p < PP; ++p) scount += ws[WS_MASK + p];
        float nll = -ws[0] / (float)(BB * TT);
        float reg = (scount / (float)PP) * (0.05f * ws[1] + 0.05f * ws[2]);
        out[0] = nll + reg;
    }
}

extern "C" void kernel_launch(void* const* d_in, const int* in_sizes, int n_in,
                              void* d_out, int out_size, void* d_ws, size_t ws_size,
                              hipStream_t stream) {
    (void)in_sizes; (void)n_in; (void)out_size; (void)ws_size;
    const float* x       = (const float*)d_in[0];
    const float* mu_pop  = (const float*)d_in[1];
    const float* L_pop   = (const float*)d_in[2];
    const float* mu_subj = (const float*)d_in[3];
    const float* L_subj  = (const float*)d_in[4];
    const float* gamma   = (const float*)d_in[5];
    const int*   sid     = (const int*)d_in[6];
    float* ws  = (float*)d_ws;
    float* out = (float*)d_out;

    hgmm_init_ws<<<1, 128, 0, stream>>>(ws);
    hgmm_mask<<<1, 128, 0, stream>>>(sid, ws);
    hgmm_main<<<BB * KK, 128, 0, stream>>>(x, mu_subj, L_subj, gamma, sid, ws);
    const int nL = PP * KK * DD * DD;
    hgmm_reg<<<(nL + 255) / 256, 256, 0, stream>>>(mu_pop, L_pop, mu_subj, L_subj, ws);
    hgmm_finalize<<<1, 32, 0, stream>>>(ws, out);
}